// QuadraticKernelMulti2D_20779051778538
// MI455X (gfx1250) — compile-verified
//
#include <hip/hip_runtime.h>
#include <hip/hip_bf16.h>

typedef __attribute__((ext_vector_type(2))) float v2f;
typedef __attribute__((ext_vector_type(8))) float v8f;

#define KSIZE   121              // 11*11 positions
#define NPAIRS  (512 * 512)      // O * I
#define PAIR_TILES (NPAIRS / 16) // 16384
#define WAVES_PER_BLOCK 8

// dists(pair, pos) = A*x1^2 + B*x1*x2 + C*x2^2, computed as a K=4 (padded) WMMA:
//   A-matrix row m = (A_m, B_m, C_m, 0), B-matrix col n = (x1n^2, x1n*x2n, x2n^2, 0)
__global__ __launch_bounds__(256)
void quad_kernel_wmma(const float* __restrict__ chol,   // (O*I, 4) row-major 2x2
                      const float* __restrict__ pos,    // (2, 121)
                      float* __restrict__ out) {        // (O*I, 121)
    const int lane = threadIdx.x & 31;
    const int wave = threadIdx.x >> 5;
    const int pairTile = blockIdx.x * WAVES_PER_BLOCK + wave; // < 16384 by grid
    const int pairBase = pairTile * 16;
    const int m  = lane & 15;
    const bool hi = lane >= 16;

    // ---- per-pair coefficients: all 32 lanes compute pair (pairBase + m)
    // (hi half is a redundant broadcast -- avoids cross-lane shuffles) ----
    const float4 cr = ((const float4*)chol)[pairBase + m]; // {a, b, c, d}
    const float inv11 = __expf(-cr.x);      // 1/l11
    const float l21   = cr.z;
    const float inv22 = __expf(-cr.w);      // 1/l22
    const float i22sq = inv22 * inv22;
    const float t     = l21 * inv11;
    const float Ac = inv11 * inv11 + t * t * i22sq;
    const float Bc = -2.0f * t * i22sq;
    const float Cc = i22sq;

    // A-matrix (16x4 f32): lanes 0-15 -> K={0,1}=(A,B); lanes 16-31 -> K={2,3}=(C,0)
    v2f Areg;
    Areg.x = hi ? Cc : Ac;
    Areg.y = hi ? 0.0f : Bc;

    const int outBase = pairBase * KSIZE;

    for (int tile = 0; tile < 8; ++tile) {
        const int col = tile * 16 + m;            // position index for column N=m
        const int cc  = col < KSIZE ? col : KSIZE - 1; // clamp OOB lanes (masked at store)
        const float x1 = pos[cc];
        const float x2 = pos[KSIZE + cc];

        // B-matrix (4x16 f32): lanes 0-15 -> K={0,1}=(x1^2, x1*x2); 16-31 -> K={2,3}=(x2^2, 0)
        v2f Breg;
        Breg.x = hi ? (x2 * x2) : (x1 * x1);
        Breg.y = hi ? 0.0f : (x1 * x2);

        v8f Creg = {};
        // 8 args: (neg_a, A, neg_b, B, c_mod, C, reuse_a, reuse_b)
        Creg = __builtin_amdgcn_wmma_f32_16x16x4_f32(
            false, Areg, false, Breg, (short)0, Creg, false, false);

        // D layout: VGPR r holds row r (lanes 0-15) / row r+8 (lanes 16-31), N = lane&15.
        // Each VGPR store = two contiguous 64B runs -> coalesced against (pair,pos) layout.
        if (col < KSIZE) {
            const int base = outBase + (hi ? 8 * KSIZE : 0) + col;
            #pragma unroll
            for (int r = 0; r < 8; ++r)
                out[base + r * KSIZE] = Creg[r];
        }
    }
}

extern "C" void kernel_launch(void* const* d_in, const int* in_sizes, int n_in,
                              void* d_out, int out_size, void* d_ws, size_t ws_size,
                              hipStream_t stream) {
    const float* chol = (const float*)d_in[0]; // (512,512,2,2) f32
    const float* pos  = (const float*)d_in[1]; // (2,121) f32
    float* out = (float*)d_out;                // (512,512,11,11) f32

    const dim3 block(256);
    const dim3 grid(PAIR_TILES / WAVES_PER_BLOCK); // 2048
    hipLaunchKernelGGL(quad_kernel_wmma, grid, block, 0, stream, chol, pos, out);
}